// Net_87875030876683
// MI455X (gfx1250) — compile-verified
//
#include <hip/hip_runtime.h>
#include <math.h>

typedef __attribute__((ext_vector_type(2))) float v2f;
typedef __attribute__((ext_vector_type(8))) float v8f;

#define NFEAT 512
#define HID   16
#define NCLS  7

// ---------------- utility ----------------
__global__ void k_zero(float* __restrict__ p, long long n) {
  long long i = (long long)blockIdx.x * blockDim.x + threadIdx.x;
  if (i < n) p[i] = 0.0f;
}

// ---------------- degree / norm ----------------
__global__ void k_degree(const long long* __restrict__ dst, float* __restrict__ deg, int E) {
  int e = blockIdx.x * blockDim.x + threadIdx.x;
  if (e < E) atomicAdd(&deg[(long long)dst[e]], 1.0f);
}

__global__ void k_dinv(const float* __restrict__ deg, float* __restrict__ dinv, int N) {
  int i = blockIdx.x * blockDim.x + threadIdx.x;
  if (i < N) dinv[i] = rsqrtf(deg[i] + 1.0f);   // +1 = self loop; always > 0
}

__global__ void k_norm(const long long* __restrict__ src, const long long* __restrict__ dst,
                       const float* __restrict__ dinv, float* __restrict__ nrm, int E) {
  int e = blockIdx.x * blockDim.x + threadIdx.x;
  if (e < E) nrm[e] = dinv[src[e]] * dinv[dst[e]];
}

// ---------------- layer-1 GEMM: h1 = x @ W1 via V_WMMA_F32_16X16X4_F32 ----------------
// One wave computes TWO 16x16 output tiles (32 rows), sharing the B fragment:
//   - two independent accumulator chains -> overlapped WMMA issue
//   - B fragments staged pre-swizzled in LDS (one ds_load_b64 per k-step, conflict-free)
// Fragment layouts per ISA 7.12.2 (32-bit):
//   A 16x4: lane L<16 holds (M=L, K=0..1); lane L+16 holds (M=L, K=2..3)
//   B 4x16: VGPR0 = {K0 (lanes 0-15), K2 (lanes 16-31)}, VGPR1 = {K1, K3}
//   C/D 16x16: VGPR v -> M = v + 8*(lane/16), N = lane%16
__global__ __launch_bounds__(256)
void k_gemm1_wmma(const float* __restrict__ x, const float* __restrict__ W1,
                  float* __restrict__ h1, int nPairs) {
  __shared__ float w1s[(NFEAT / 4) * 64];            // 128 k-steps x 32 lanes x 2 floats = 32KB

  // cooperative pre-swizzle of W1 into WMMA B-fragment order
  for (int idx = threadIdx.x; idx < (NFEAT / 4) * 32; idx += 256) {
    int k = idx >> 5, l = idx & 31;
    int lh = l >> 4, lm = l & 15;
    int kk = k * 4 + lh * 2;
    w1s[idx * 2 + 0] = W1[kk * HID + lm];            // (K=kk,   N=lm)
    w1s[idx * 2 + 1] = W1[(kk + 1) * HID + lm];      // (K=kk+1, N=lm)
  }
  __syncthreads();

  int pair = (int)((blockIdx.x * blockDim.x + threadIdx.x) >> 5);
  if (pair >= nPairs) return;                        // wave-uniform: EXEC stays all-ones

  const int lane = threadIdx.x & 31;
  const int half = lane >> 4;
  const int m    = lane & 15;
  const long long r0 = (long long)pair * 32;

  const float* xrow0 = x + (r0 + m) * NFEAT + half * 2;       // rows r0 .. r0+15
  const float* xrow1 = xrow0 + 16LL * NFEAT;                  // rows r0+16 .. r0+31
  const float* bsrc  = w1s + lane * 2;

  v8f acc0 = {}, acc1 = {};

#pragma unroll 4
  for (int k = 0; k < NFEAT / 4; ++k) {
    v2f b  = *(const v2f*)(bsrc + (k << 6));                  // ds_load_b64, no conflicts
    v2f a0 = *(const v2f*)(xrow0 + k * 4);                    // global b64
    v2f a1 = *(const v2f*)(xrow1 + k * 4);
    acc0 = __builtin_amdgcn_wmma_f32_16x16x4_f32(false, a0, false, b, (short)0, acc0, false, false);
    acc1 = __builtin_amdgcn_wmma_f32_16x16x4_f32(false, a1, false, b, (short)0, acc1, false, false);
  }

  float* orow0 = h1 + r0 * HID + m;
  float* orow1 = orow0 + 16LL * HID;
#pragma unroll
  for (int v = 0; v < 8; ++v) {
    orow0[(long long)(v + half * 8) * HID] = acc0[v];
    orow1[(long long)(v + half * 8) * HID] = acc1[v];
  }
}

// ---------------- edge scatter (16 feats) ----------------
__global__ void k_scatter16(const long long* __restrict__ src, const long long* __restrict__ dst,
                            const float* __restrict__ nrmv, const float* __restrict__ h,
                            float* __restrict__ agg, int E) {
  int e = blockIdx.x * blockDim.x + threadIdx.x;
  if (e >= E) return;
  long long s = src[e], d = dst[e];
  float nrm = nrmv[e];
  const float4* hp = (const float4*)(h + s * HID);
  float* ap = agg + d * HID;
#pragma unroll
  for (int i = 0; i < 4; ++i) {
    float4 v = hp[i];
    atomicAdd(ap + i * 4 + 0, nrm * v.x);
    atomicAdd(ap + i * 4 + 1, nrm * v.y);
    atomicAdd(ap + i * 4 + 2, nrm * v.z);
    atomicAdd(ap + i * 4 + 3, nrm * v.w);
  }
}

// ---------------- self-loop + bias + ReLU (in place on agg1) ----------------
__global__ void k_final1(const float* __restrict__ h1, const float* __restrict__ b1,
                         const float* __restrict__ dinv, float* __restrict__ agg, int N) {
  int i = blockIdx.x * blockDim.x + threadIdx.x;
  if (i >= N) return;
  float sl = dinv[i] * dinv[i];
  long long base = (long long)i * HID;
#pragma unroll
  for (int f = 0; f < HID; ++f) {
    float v = agg[base + f] + sl * h1[base + f] + b1[f];
    agg[base + f] = v > 0.0f ? v : 0.0f;
  }
}

// ---------------- layer-2 GEMM: h2 = hact @ W2 (16x7, W2 in LDS) ----------------
__global__ void k_gemm2(const float* __restrict__ hact, const float* __restrict__ W2,
                        float* __restrict__ h2, int N) {
  __shared__ float w[HID * NCLS];
  if (threadIdx.x < HID * NCLS) w[threadIdx.x] = W2[threadIdx.x];
  __syncthreads();
  int i = blockIdx.x * blockDim.x + threadIdx.x;
  if (i >= N) return;
  float r[HID];
  const float4* hp = (const float4*)(hact + (long long)i * HID);
#pragma unroll
  for (int q = 0; q < 4; ++q) {
    float4 v = hp[q];
    r[q * 4 + 0] = v.x; r[q * 4 + 1] = v.y; r[q * 4 + 2] = v.z; r[q * 4 + 3] = v.w;
  }
  float acc[NCLS] = {0.f, 0.f, 0.f, 0.f, 0.f, 0.f, 0.f};
#pragma unroll
  for (int f = 0; f < HID; ++f)
#pragma unroll
    for (int c = 0; c < NCLS; ++c)
      acc[c] += r[f] * w[f * NCLS + c];
  float* o = h2 + (long long)i * 8;
#pragma unroll
  for (int c = 0; c < NCLS; ++c) o[c] = acc[c];
  o[7] = 0.0f;
}

// ---------------- edge scatter (7 feats, stride 8) ----------------
__global__ void k_scatter8(const long long* __restrict__ src, const long long* __restrict__ dst,
                           const float* __restrict__ nrmv, const float* __restrict__ h,
                           float* __restrict__ agg, int E) {
  int e = blockIdx.x * blockDim.x + threadIdx.x;
  if (e >= E) return;
  long long s = src[e], d = dst[e];
  float nrm = nrmv[e];
  const float4* hp = (const float4*)(h + s * 8);
  float4 a = hp[0], b = hp[1];
  float* ap = agg + d * 8;
  atomicAdd(ap + 0, nrm * a.x);
  atomicAdd(ap + 1, nrm * a.y);
  atomicAdd(ap + 2, nrm * a.z);
  atomicAdd(ap + 3, nrm * a.w);
  atomicAdd(ap + 4, nrm * b.x);
  atomicAdd(ap + 5, nrm * b.y);
  atomicAdd(ap + 6, nrm * b.z);
}

// ---------------- self-loop + bias + log_softmax ----------------
__global__ void k_final2(const float* __restrict__ h2, const float* __restrict__ b2,
                         const float* __restrict__ dinv, const float* __restrict__ agg,
                         float* __restrict__ out, int N) {
  int i = blockIdx.x * blockDim.x + threadIdx.x;
  if (i >= N) return;
  float sl = dinv[i] * dinv[i];
  long long base = (long long)i * 8;
  float l[NCLS];
  float mx = -INFINITY;
#pragma unroll
  for (int c = 0; c < NCLS; ++c) {
    l[c] = agg[base + c] + sl * h2[base + c] + b2[c];
    mx = fmaxf(mx, l[c]);
  }
  float sum = 0.0f;
#pragma unroll
  for (int c = 0; c < NCLS; ++c) sum += expf(l[c] - mx);
  float lse = logf(sum);
  float* o = out + (long long)i * NCLS;
#pragma unroll
  for (int c = 0; c < NCLS; ++c) o[c] = l[c] - mx - lse;
}

// ---------------- driver ----------------
extern "C" void kernel_launch(void* const* d_in, const int* in_sizes, int n_in,
                              void* d_out, int out_size, void* d_ws, size_t ws_size,
                              hipStream_t stream) {
  const float*      x   = (const float*)d_in[0];
  const long long*  ei  = (const long long*)d_in[1];
  const float*      W1  = (const float*)d_in[2];
  const float*      b1  = (const float*)d_in[3];
  const float*      W2  = (const float*)d_in[4];
  const float*      b2  = (const float*)d_in[5];
  float*            out = (float*)d_out;

  const int N = in_sizes[0] / NFEAT;     // 100000
  const int E = in_sizes[1] / 2;         // 3200000
  const long long* src = ei;
  const long long* dst = ei + E;

  // workspace carve-up (floats)
  long long pad = ((long long)N + 255) & ~255LL;
  float* w    = (float*)d_ws;
  float* deg  = w;
  float* dinv = deg  + pad;
  float* h1   = dinv + pad;
  float* agg1 = h1   + (long long)N * HID;   // later: ReLU'd activations in place
  float* h2   = agg1 + (long long)N * HID;
  float* agg2 = h2   + (long long)N * 8;
  float* nrm  = agg2 + (long long)N * 8;     // E floats

  const int T = 256;
  dim3 b(T);
  auto blocks = [](long long n, int t) { return dim3((unsigned)((n + t - 1) / t)); };

  // zero accumulators (every call: graph-replay safe)
  k_zero<<<blocks(N, T), b, 0, stream>>>(deg,  N);
  k_zero<<<blocks((long long)N * HID, T), b, 0, stream>>>(agg1, (long long)N * HID);
  k_zero<<<blocks((long long)N * 8,   T), b, 0, stream>>>(agg2, (long long)N * 8);

  // degrees + normalization
  k_degree<<<blocks(E, T), b, 0, stream>>>(dst, deg, E);
  k_dinv<<<blocks(N, T), b, 0, stream>>>(deg, dinv, N);
  k_norm<<<blocks(E, T), b, 0, stream>>>(src, dst, dinv, nrm, E);

  // layer 1
  int nPairs = N / 32;                                      // 3125 wave jobs, 2 tiles each
  k_gemm1_wmma<<<blocks((long long)nPairs * 32, T), b, 0, stream>>>(x, W1, h1, nPairs);
  k_scatter16<<<blocks(E, T), b, 0, stream>>>(src, dst, nrm, h1, agg1, E);
  k_final1<<<blocks(N, T), b, 0, stream>>>(h1, b1, dinv, agg1, N);

  // layer 2
  k_gemm2<<<blocks(N, T), b, 0, stream>>>(agg1, W2, h2, N);
  k_scatter8<<<blocks(E, T), b, 0, stream>>>(src, dst, nrm, h2, agg2, E);
  k_final2<<<blocks(N, T), b, 0, stream>>>(h2, b2, dinv, agg2, out, N);
}